// GAttentionLayer_7791070675887
// MI455X (gfx1250) — compile-verified
//
#include <hip/hip_runtime.h>

typedef float v2f __attribute__((ext_vector_type(2)));
typedef float v4f __attribute__((ext_vector_type(4)));
typedef float v8f __attribute__((ext_vector_type(8)));

#define NNODE 6
#define DIM   64
#define WAVES 8
#define BPW   8               // batches per wave
#define ROWS  (BPW*NNODE)     // 48 activation rows per wave
#define XST   68              // padded LDS row stride (floats): 16B-aligned, bank-friendly
#define XWSZ  (ROWS*XST)      // 3264 floats per wave

__global__ __launch_bounds__(256)
void gat_wmma_kernel(const float* __restrict__ node,
                     const int*   __restrict__ adj,
                     const float* __restrict__ W0, const float* __restrict__ b0,
                     const float* __restrict__ W1, const float* __restrict__ b1,
                     const float* __restrict__ W2, const float* __restrict__ b2,
                     const float* __restrict__ avec,
                     float* __restrict__ out)
{
    __shared__ float s_w[3*4096];          // packed weights: [l][(k>>1)*128 + n*2 + (k&1)] = W_l[n][k]
    __shared__ float s_b[3*64];
    __shared__ float s_a[128];
    __shared__ int   s_adj[36];
    __shared__ float s_x[WAVES*XWSZ];      // per-wave activation buffer (row-major, stride XST)
    __shared__ float s_e[WAVES*ROWS*2];    // per-row {e_src, e_dst}

    const int tid  = threadIdx.x;
    const int wave = tid >> 5;
    const int lane = tid & 31;
    const int li   = lane & 15;
    const int hi   = lane >> 4;

    // ---------------- stage shared weights / vectors ----------------
    {
        const float* Ws[3] = {W0, W1, W2};
#pragma unroll
        for (int l = 0; l < 3; ++l) {
            const float* W = Ws[l];
#pragma unroll
            for (int it = 0; it < 4; ++it) {
                int f = tid*4 + it*1024;          // 0..4095 over the 64x64 matrix
                int n = f >> 6, k = f & 63;       // k multiple of 4
                v4f wv = *(const v4f*)(W + n*64 + k);
                int base = l*4096 + (k>>1)*128 + n*2;
                s_w[base]     = wv.x;             // k
                s_w[base+1]   = wv.y;             // k+1
                s_w[base+128] = wv.z;             // k+2
                s_w[base+129] = wv.w;             // k+3
            }
        }
        if (tid < 64) {
            s_b[tid]      = b0[tid];
            s_b[64+tid]   = b1[tid];
            s_b[128+tid]  = b2[tid];
        }
        if (tid < 128) s_a[tid]  = avec[tid];
        if (tid < 36)  s_adj[tid] = adj[tid];
    }

    // ---------------- stage this wave's node rows ----------------
    float* xw = &s_x[wave*XWSZ];
    {
        const float* np = node + ((size_t)blockIdx.x*64 + (size_t)wave*BPW) * (NNODE*DIM);
#pragma unroll
        for (int it = 0; it < 24; ++it) {
            int f = lane*4 + it*128;              // 0..3071
            int r = f >> 6, c = f & 63;
            v4f v = *(const v4f*)(np + f);
            *(v4f*)&xw[r*XST + c] = v;
        }
    }
    __syncthreads();

    // ---------------- 3 chained GEMMs via V_WMMA_F32_16X16X4_F32 ----------------
    // out[m][n] = sum_k X[m][k] * W[n][k]  (A = X, B = W^T)
    for (int l = 0; l < 3; ++l) {
        for (int mt = 0; mt < 3; ++mt) {
            // A fragments (ISA 16x4 fp32 layout): lane holds row (mt*16+li),
            // k-pair base = 4*kk + 2*hi
            v2f afrag[16];
            const int arow = mt*16 + li;
#pragma unroll
            for (int kk = 0; kk < 16; ++kk) {
                int kb = kk*4 + hi*2;
                afrag[kk] = *(const v2f*)&xw[arow*XST + kb];
            }
            for (int nt = 0; nt < 4; ++nt) {
                const int n = nt*16 + li;
                v8f acc = {};
#pragma unroll
                for (int kk = 0; kk < 16; ++kk) {
                    int kp = kk*2 + hi;           // = (4*kk + 2*hi)/2
                    v2f bfrag = *(const v2f*)&s_w[l*4096 + kp*128 + n*2];
                    acc = __builtin_amdgcn_wmma_f32_16x16x4_f32(
                              false, afrag[kk], false, bfrag,
                              (short)0, acc, false, false);
                }
                const float bias = s_b[l*64 + n];
                // D layout: lane = n%16 + 16*(m/8), vgpr v = m%8
#pragma unroll
                for (int v = 0; v < 8; ++v) {
                    int drow = mt*16 + hi*8 + v;
                    xw[drow*XST + n] = acc[v] + bias;
                }
            }
        }
        __syncthreads();
    }

    // ---------------- per-row attention scalars e_src / e_dst ----------------
    float* ew = &s_e[wave*ROWS*2];
    for (int ch = 0; ch < 2; ++ch) {
        int r = ch*32 + lane;
        if (r < ROWS) {
            float es = 0.f, ed = 0.f;
#pragma unroll
            for (int c = 0; c < 64; c += 4) {
                v4f xv = *(const v4f*)&xw[r*XST + c];
                es += xv.x*s_a[c]    + xv.y*s_a[c+1]    + xv.z*s_a[c+2]    + xv.w*s_a[c+3];
                ed += xv.x*s_a[64+c] + xv.y*s_a[64+c+1] + xv.z*s_a[64+c+2] + xv.w*s_a[64+c+3];
            }
            ew[r*2]   = es;
            ew[r*2+1] = ed;
        }
    }
    __syncthreads();

    // ---------------- masked softmax + aggregation (4 lanes per batch) ----------------
    {
        const int g   = lane >> 2;    // batch within wave (0..7)
        const int sub = lane & 3;     // 16-column slice
        float es6[6], ed6[6];
#pragma unroll
        for (int j = 0; j < 6; ++j) {
            es6[j] = ew[(g*6+j)*2];
            ed6[j] = ew[(g*6+j)*2+1];
        }
        float p[36];
#pragma unroll
        for (int i = 0; i < 6; ++i) {
            float m = -3.0e38f;
#pragma unroll
            for (int j = 0; j < 6; ++j) {
                float z = es6[i] + ed6[j];
                z = (z >= 0.f) ? z : 0.2f*z;                 // LeakyReLU(0.2)
                z = (s_adj[i*6+j] == 1) ? z : -9.0e15f;      // adjacency mask
                p[i*6+j] = z;
                m = fmaxf(m, z);
            }
            float sum = 0.f;
#pragma unroll
            for (int j = 0; j < 6; ++j) {
                float e = __expf(p[i*6+j] - m);
                p[i*6+j] = e;
                sum += e;
            }
            float inv = 1.0f / sum;
#pragma unroll
            for (int j = 0; j < 6; ++j) p[i*6+j] *= inv;
        }
        const size_t bb = (size_t)blockIdx.x*64 + (size_t)wave*BPW + g;
        float* op = out + bb*(NNODE*DIM);
#pragma unroll
        for (int ch = 0; ch < 4; ++ch) {
            const int col = sub*16 + ch*4;
            v4f accv[6];
#pragma unroll
            for (int i = 0; i < 6; ++i) { v4f z = {}; accv[i] = z; }
#pragma unroll
            for (int j = 0; j < 6; ++j) {
                v4f xv = *(const v4f*)&xw[(g*6+j)*XST + col];
#pragma unroll
                for (int i = 0; i < 6; ++i)
                    accv[i] += p[i*6+j] * xv;
            }
#pragma unroll
            for (int i = 0; i < 6; ++i)
                *(v4f*)(op + i*64 + col) = accv[i];
        }
    }
}

extern "C" void kernel_launch(void* const* d_in, const int* in_sizes, int n_in,
                              void* d_out, int out_size, void* d_ws, size_t ws_size,
                              hipStream_t stream) {
    (void)in_sizes; (void)n_in; (void)out_size; (void)d_ws; (void)ws_size;
    const float* node = (const float*)d_in[0];
    const int*   adj  = (const int*)  d_in[1];
    const float* W0   = (const float*)d_in[2];
    const float* b0   = (const float*)d_in[3];
    const float* W1   = (const float*)d_in[4];
    const float* b1   = (const float*)d_in[5];
    const float* W2   = (const float*)d_in[6];
    const float* b2   = (const float*)d_in[7];
    const float* avec = (const float*)d_in[8];
    float* out = (float*)d_out;

    // 262144 batches / 64 batches per block = 4096 blocks of 256 threads (8 wave32)
    dim3 grid(4096), block(256);
    gat_wmma_kernel<<<grid, block, 0, stream>>>(node, adj, W0, b0, W1, b1, W2, b2, avec, out);
}